// InterpolationBlock1D_Lin_26010321944829
// MI455X (gfx1250) — compile-verified
//
#include <hip/hip_runtime.h>
#include <hip/hip_bf16.h>
#include <stdint.h>

typedef __attribute__((ext_vector_type(4))) float f4;
typedef __attribute__((ext_vector_type(2))) int   i2;

#define WAVES_PER_BLOCK 4
#define BLOCK_THREADS   (WAVES_PER_BLOCK * 32)
#define ROW_WORDS       36      // 32 data words + 4 pad words (16B) -> breaks 128B LDS bank stride
#define CHUNK           32      // evals per wave per pipeline stage
#define K               16

// prod[e,k] = sf[e,0,k]*nodal[conn[cid[e]].x-1] + sf[e,1,k]*nodal[conn[cid[e]].y-1]
__global__ __launch_bounds__(BLOCK_THREADS)
void interp1d_lin_kernel(const int* __restrict__ cell_id,
                         const float* __restrict__ nodal,
                         const float* __restrict__ sf,
                         const int* __restrict__ conn,
                         float* __restrict__ out,
                         int n_eval)
{
    // [parity][wave-in-block][32 evals * 36 words]  = 36,864 bytes
    __shared__ alignas(16) float lds_sf[2][WAVES_PER_BLOCK][CHUNK * ROW_WORDS];

    const int lane   = threadIdx.x & 31;
    const int wib    = threadIdx.x >> 5;
    const int wave   = blockIdx.x * WAVES_PER_BLOCK + wib;
    const int nwaves = gridDim.x * WAVES_PER_BLOCK;
    const int nchunk = (n_eval + CHUNK - 1) / CHUNK;

    if (wave >= nchunk) return;

    // LDS byte offsets of this wave's two staging buffers (generic addr low 32 bits == LDS offset)
    const unsigned lds_off0 = (unsigned)(uintptr_t)&lds_sf[0][wib][0];
    const unsigned lds_off1 = (unsigned)(uintptr_t)&lds_sf[1][wib][0];

    // Issue 8 async B128 copies: lane L stages eval (c*32+L)'s 128B row into its padded LDS row.
    // NT hint: the sf stream is consumed once from LDS -- keep it from evicting the nodal
    // gather table (4MB, re-referenced 4M times) out of the 192MB L2.
    auto issue = [&](int c, int parity) {
        const int e = min(c * CHUNK + lane, n_eval - 1);
        const unsigned long long src =
            (unsigned long long)(uintptr_t)(sf + (size_t)e * (2 * K));
        const unsigned dst =
            (parity ? lds_off1 : lds_off0) + (unsigned)lane * (ROW_WORDS * 4);
#pragma unroll
        for (int s = 0; s < 8; ++s) {
            asm volatile("global_load_async_to_lds_b128 %0, %1, off th:TH_LOAD_NT"
                         :: "v"(dst + (unsigned)(s * 16)),
                            "v"(src + (unsigned long long)(s * 16))
                         : "memory");
        }
    };

    int parity = 0;
    issue(wave, parity);                       // prologue: stage first chunk

    for (int c = wave; c < nchunk; c += nwaves) {
        const int  cn  = c + nwaves;
        const bool pre = (cn < nchunk);
        if (pre) issue(cn, parity ^ 1);        // stage next chunk into other buffer

        // Index/gather chain overlaps the in-flight async copies.
        // cell_id / connectivity are single-pass streams -> NT; nodal stays cached (RT).
        const int   e     = min(c * CHUNK + lane, n_eval - 1);
        const int   cid   = __builtin_nontemporal_load(cell_id + e);
        const i2    cpair = __builtin_nontemporal_load((const i2*)(conn + (size_t)cid * 2));
        const float n1    = nodal[cpair.x - 1];
        const float n2    = nodal[cpair.y - 1];

        // Wait for *current* chunk's 8 copies; next chunk's 8 stay outstanding.
        if (pre) asm volatile("s_wait_asynccnt 8" ::: "memory");
        else     asm volatile("s_wait_asynccnt 0" ::: "memory");

        const f4* row = (const f4*)&lds_sf[parity][wib][lane * ROW_WORDS];
        f4* o = (f4*)(out + (size_t)e * K);
#pragma unroll
        for (int j = 0; j < 4; ++j) {
            const f4 a = row[j];       // sf[e,0, 4j..4j+3]
            const f4 b = row[j + 4];   // sf[e,1, 4j..4j+3]
            const f4 r = a * n1 + b * n2;
            __builtin_nontemporal_store(r, o + j);  // written once; keep out of L2
        }
        parity ^= 1;
    }
}

extern "C" void kernel_launch(void* const* d_in, const int* in_sizes, int n_in,
                              void* d_out, int out_size, void* d_ws, size_t ws_size,
                              hipStream_t stream) {
    // setup_inputs() order: x, cell_id, nodal_values, shape_functions, connectivity
    const int*   cell_id = (const int*)d_in[1];
    const float* nodal   = (const float*)d_in[2];
    const float* sf      = (const float*)d_in[3];
    const int*   conn    = (const int*)d_in[4];
    float*       out     = (float*)d_out;

    const int n_eval = in_sizes[1];

    const int blocks = 2048;   // 8192 waves -> ~7.6 pipelined chunks per wave
    interp1d_lin_kernel<<<blocks, BLOCK_THREADS, 0, stream>>>(
        cell_id, nodal, sf, conn, out, n_eval);
}